// OrgLieSE3_66649302499846
// MI455X (gfx1250) — compile-verified
//
#include <hip/hip_runtime.h>

// SE(2)-embedded-in-SE(3) exp map, fully closed form (z-axis rotation only).
// B = 4,194,304 elements: 48 MB in, 256 MB out -> pure HBM-streaming kernel
// (~13 us floor at 23.3 TB/s). One thread per OUTPUT ROW (4 threads per
// matrix) so stores are perfectly coalesced global_store_b128 (512B
// contiguous per wave-instruction); quad lanes cooperatively load the 3
// input components (one dense b32 each) and broadcast via wave32 shuffle
// (ds_bpermute_b32). Divisions replaced by hardware v_rcp_f32 to keep
// VALU-per-byte below the bandwidth-saturation threshold.

#define ROTATION_SCALE    0.1f
#define EPSV              1e-5f

__global__ __launch_bounds__(256) void
OrgLieSE3_rows_kernel(const float* __restrict__ uv,
                      float* __restrict__ out,
                      int B) {
    const int tid = blockIdx.x * blockDim.x + threadIdx.x;
    const int m = tid >> 2;        // matrix index (shared by a quad of lanes)
    const int r = tid & 3;         // output row 0..3
    if (m >= B) return;

    // ---- cooperative input load: lane r of the quad fetches component r (lane 3 refetches 0)
    const int lane  = threadIdx.x & 31;
    const int qbase = lane & ~3;                 // first lane of this quad
    const int comp  = (r < 3) ? r : 0;
    const size_t inIdx = (size_t)m * 3 + (size_t)comp;

    // prefetch the input stream ~8KB ahead (speculative; OOB is dropped)
    __builtin_prefetch(uv + inIdx + 2048, 0, 0);

    const float v = uv[inIdx];

    // broadcast the 3 components to all 4 lanes of the quad (ds_bpermute, wave32)
    const float x = __shfl(v, qbase + 0, 32);                    // *1.0 (TS)
    const float y = __shfl(v, qbase + 1, 32);                    // *1.0 (TS)
    const float w = __shfl(v, qbase + 2, 32) * ROTATION_SCALE;

    // ---- closed-form coefficients (2x v_rcp_f32 instead of 3 IEEE divides)
    const float t2 = w * w;            // theta^2
    const float th = fabsf(w);         // theta = sqrt(w^2)
    const float s  = __sinf(th);
    const float c  = __cosf(th);
    const float rth = __builtin_amdgcn_rcpf(th + EPSV);   // 1/(theta+eps)
    const float rt2 = __builtin_amdgcn_rcpf(t2 + EPSV);   // 1/(theta^2+eps)
    const float A  = s * rth;                   // sin(th)/(th+eps)
    const float Bc = (1.0f - c) * rt2;          // (1-cos)/(th^2+eps)
    const float C  = (1.0f - A) * rt2;          // (1-A)/(th^2+eps)

    // R = I + skew*A + skew^2*B ; skew^2 = diag(-w^2, -w^2, 0)
    const float r00 = fmaf(-t2, Bc, 1.0f);             // = r11
    const float wA  = w * A;
    // V = I + skew*B + skew^2*C
    const float v00 = fmaf(-t2, C, 1.0f);              // = v11
    const float wB  = w * Bc;
    // t = V @ (x, y, 0)
    const float tx  = fmaf(v00, x, -(wB * y));
    const float ty  = fmaf(wB,  x,  v00 * y);

    // ---- branchless row select
    //  r0: [ r00, -wA, 0, tx ]
    //  r1: [  wA, r00, 0, ty ]
    //  r2: [   0,   0, 1, 0  ]
    //  r3: [   0,   0, 0, 1  ]
    float4 row;
    row.x = (r == 0) ?  r00 : ((r == 1) ? wA  : 0.0f);
    row.y = (r == 0) ? -wA  : ((r == 1) ? r00 : 0.0f);
    row.z = (r == 2) ? 1.0f : 0.0f;
    row.w = (r == 0) ?  tx  : ((r == 1) ? ty  : ((r == 3) ? 1.0f : 0.0f));

    // ---- perfectly coalesced b128 store: consecutive lanes -> consecutive 16B
    reinterpret_cast<float4*>(out)[tid] = row;
}

extern "C" void kernel_launch(void* const* d_in, const int* in_sizes, int n_in,
                              void* d_out, int out_size, void* d_ws, size_t ws_size,
                              hipStream_t stream) {
    const float* uv  = (const float*)d_in[0];
    float*       out = (float*)d_out;

    const int B    = in_sizes[0] / 3;      // 4,194,304
    const int rows = B * 4;                // one thread per output row
    const int block = 256;
    const int grid  = (rows + block - 1) / block;

    OrgLieSE3_rows_kernel<<<grid, block, 0, stream>>>(uv, out, B);
}